// CausalSelfAttention_45801531244943
// MI455X (gfx1250) — compile-verified
//
#include <hip/hip_runtime.h>
#include <hip/hip_bf16.h>

// ---------------------------------------------------------------------------
// CDNA5 (gfx1250) fp32 WMMA implementation of windowed+prefix causal attention
//   x[B,S,1024] @ w_qkv^T -> Q,K,V [B,H,S,64]
//   flash-style attention (WINDOW=512 local + PREFIX=16 global, causal)
//   O[B,S,1024] @ w_out^T -> out
// All matrix math on V_WMMA_F32_16X16X4_F32 (exact fp32, matches reference).
// ---------------------------------------------------------------------------

typedef float v2f __attribute__((ext_vector_type(2)));
typedef float v8f __attribute__((ext_vector_type(8)));

#define WMMA_F32(a, b, c) \
  __builtin_amdgcn_wmma_f32_16x16x4_f32(false, (a), false, (b), (short)0, (c), false, false)

// ---------------------------------------------------------------------------
// NT GEMM: C[m,n] = sum_k A[m,k] * Bw[n,k]
//   A  : [M,K] row-major,  Bw : [N,K] row-major (both K-contiguous)
// Block: 256 threads = 8 waves, tile 128(M) x 64(N), BK=32.
// Each wave computes a 16(M) x 64(N) strip = 4 WMMA accumulators, so each A
// fragment is reused by 4 WMMAs (1.25 LDS b64 loads per WMMA instead of 2).
// MODE 0: plain store C[M,N] row-major.
// MODE 1: QKV scatter -> Cout is base of workspace [3][B=2][H=16][S=4096][64].
// ---------------------------------------------------------------------------
template <int MODE>
__global__ __launch_bounds__(256) void gemm_nt_wmma(const float* __restrict__ A,
                                                    const float* __restrict__ Bw,
                                                    float* __restrict__ Cout,
                                                    int M, int N, int K) {
  constexpr int LDA = 36;  // padded LDS row stride (dwords): 36*4 bytes = 16B-aligned rows
  __shared__ float As[128 * LDA];
  __shared__ float Bs[64 * LDA];

  const int tid  = threadIdx.x;
  const int lane = tid & 31;
  const int wave = tid >> 5;      // 0..7 -> M strip
  const int m0   = blockIdx.y * 128;
  const int n0   = blockIdx.x * 64;

  const int lr   = lane & 15;           // row within A strip / col within B tile
  const int koff = (lane >> 4) << 1;    // K pair: lanes 0-15 -> K{0,1}, 16-31 -> K{2,3}

  // cooperative-load indices: A = 4 float4 per thread, B = 2 float4 per thread
  const int la_r = tid >> 3;            // 0..31
  const int la_k = (tid & 7) << 2;      // 0,4,...,28

  v8f acc0 = {}, acc1 = {}, acc2 = {}, acc3 = {};

  for (int k0 = 0; k0 < K; k0 += 32) {
    __syncthreads();
    {
#pragma unroll
      for (int r4 = 0; r4 < 4; ++r4) {
        const int r = la_r + 32 * r4;
        const float4 va = *(const float4*)(A + (size_t)(m0 + r) * K + k0 + la_k);
        *(float4*)&As[r * LDA + la_k] = va;
      }
      const float4 vb0 = *(const float4*)(Bw + (size_t)(n0 + la_r) * K + k0 + la_k);
      *(float4*)&Bs[la_r * LDA + la_k] = vb0;
      const float4 vb1 = *(const float4*)(Bw + (size_t)(n0 + 32 + la_r) * K + k0 + la_k);
      *(float4*)&Bs[(32 + la_r) * LDA + la_k] = vb1;
    }
    __syncthreads();
#pragma unroll
    for (int j = 0; j < 8; ++j) {
      // A operand: lane = M row (lane%16), vgpr pair = K = 4j + koff + {0,1}
      v2f a = *(const v2f*)&As[(wave * 16 + lr) * LDA + j * 4 + koff];
      // B operand: lane = N col, same K pair; Bw stored [n][k] so read is contiguous
      v2f b0 = *(const v2f*)&Bs[(0 * 16 + lr) * LDA + j * 4 + koff];
      acc0 = WMMA_F32(a, b0, acc0);
      v2f b1 = *(const v2f*)&Bs[(1 * 16 + lr) * LDA + j * 4 + koff];
      acc1 = WMMA_F32(a, b1, acc1);
      v2f b2 = *(const v2f*)&Bs[(2 * 16 + lr) * LDA + j * 4 + koff];
      acc2 = WMMA_F32(a, b2, acc2);
      v2f b3 = *(const v2f*)&Bs[(3 * 16 + lr) * LDA + j * 4 + koff];
      acc3 = WMMA_F32(a, b3, acc3);
    }
  }

  // C/D layout: vgpr i, lanes 0-15 -> row i, col lane%16; lanes 16-31 -> row i+8
  const int rbase = m0 + wave * 16 + ((lane >> 4) << 3);

  if (MODE == 0) {
#pragma unroll
    for (int nc = 0; nc < 4; ++nc) {
      const int col = n0 + nc * 16 + lr;
      const v8f acc = (nc == 0) ? acc0 : (nc == 1) ? acc1 : (nc == 2) ? acc2 : acc3;
#pragma unroll
      for (int i = 0; i < 8; ++i)
        Cout[(size_t)(rbase + i) * N + col] = acc[i];
    }
  } else {
#pragma unroll
    for (int nc = 0; nc < 4; ++nc) {
      const int col = n0 + nc * 16 + lr;
      const v8f acc = (nc == 0) ? acc0 : (nc == 1) ? acc1 : (nc == 2) ? acc2 : acc3;
      // column -> (tensor, head, dim); row -> (batch, seq)
      const int tsel = col >> 10;        // 0=Q 1=K 2=V
      const int r1   = col & 1023;
      const int h    = r1 >> 6;
      const int d    = r1 & 63;
      float* dst = Cout + (size_t)tsel * (2ull * 16 * 4096 * 64);
#pragma unroll
      for (int i = 0; i < 8; ++i) {
        const int m = rbase + i;
        const int b = m >> 12;           // / 4096
        const int s = m & 4095;
        dst[((size_t)((b * 16 + h) * 4096 + s)) * 64 + d] = acc[i];
      }
    }
  }
}

// ---------------------------------------------------------------------------
// Attention: grid = B*H*(S/64) blocks of 128 threads (4 waves).
// Each wave owns a 16-query tile; the 4 waves share K/V tiles in LDS.
// Tile sequence (block-uniform): prefix tile t=0 (mask k<=q), then local tiles
// t in [max(1, ct-32), ct+3] (mask 0 <= q-k < 512; k>=16 implied by t>=1).
// Online softmax with per-row running max/denominator held in registers
// (row i+8*(lane/16) lives in vgpr i; row reductions = shfl_xor over 16 lanes).
// ---------------------------------------------------------------------------
__global__ __launch_bounds__(128) void attn_kernel(const float* __restrict__ Qg,
                                                   const float* __restrict__ Kg,
                                                   const float* __restrict__ Vg,
                                                   float* __restrict__ Og) {
  constexpr int S_ = 4096, H = 16, Dh = 64, WND = 512;
  constexpr int LK = 68;  // K-tile LDS row stride (dwords)
  constexpr int LV = 18;  // V^T / P LDS row stride (dwords)

  const int blk  = blockIdx.x;
  const int cq64 = blk & ((S_ / 64) - 1);   // 64-query chunk in sequence
  const int bh   = blk / (S_ / 64);         // b*H + h

  const float* Qb = Qg + (size_t)bh * S_ * Dh;
  const float* Kb = Kg + (size_t)bh * S_ * Dh;
  const float* Vb = Vg + (size_t)bh * S_ * Dh;

  const int tid  = threadIdx.x;
  const int lane = tid & 31;
  const int wave = tid >> 5;   // 0..3
  const int lr   = lane & 15;
  const int hi   = lane >> 4;
  const int koff = hi << 1;

  const int q0b = cq64 * 64;
  const int q0w = q0b + wave * 16;

  __shared__ float Ks[16 * LK];        // K tile, [key][d]
  __shared__ float VTs[64 * LV];       // V tile transposed, [d][key]
  __shared__ float Ps[4][16 * LV];     // per-wave P tile, [q][key]

  // Preload Q tile into A-operand registers: qa[j] = Q[m][4j+koff .. +1]
  v2f qa[16];
  {
    const float* qrow = Qb + (size_t)(q0w + lr) * Dh;
#pragma unroll
    for (int j = 0; j < 16; ++j)
      qa[j] = *(const v2f*)(qrow + j * 4 + koff);
  }

  v8f acc0 = {}, acc1 = {}, acc2 = {}, acc3 = {};
  float mrow[8], lrow[8];
#pragma unroll
  for (int i = 0; i < 8; ++i) { mrow[i] = -1e30f; lrow[i] = 0.0f; }

  const int ct   = q0b >> 4;
  const int t_lo = (ct - 32) > 1 ? (ct - 32) : 1;
  const int t_hi = ct + 3;  // inclusive

  for (int ti = -1; ti <= t_hi - t_lo; ++ti) {
    const int t = (ti < 0) ? 0 : (t_lo + ti);

    __syncthreads();
    {  // cooperative load: K tile 16x64 row-major, V tile transposed to [d][key]
      const int r = tid >> 4;            // 0..7 (also handles r+8)
      const int c = (tid & 15) << 2;     // 0..60 step 4
      const float4 kv0 = *(const float4*)(Kb + (size_t)(t * 16 + r) * Dh + c);
      *(float4*)&Ks[r * LK + c] = kv0;
      const float4 kv1 = *(const float4*)(Kb + (size_t)(t * 16 + r + 8) * Dh + c);
      *(float4*)&Ks[(r + 8) * LK + c] = kv1;
      const float4 vv0 = *(const float4*)(Vb + (size_t)(t * 16 + r) * Dh + c);
      VTs[(c + 0) * LV + r] = vv0.x;
      VTs[(c + 1) * LV + r] = vv0.y;
      VTs[(c + 2) * LV + r] = vv0.z;
      VTs[(c + 3) * LV + r] = vv0.w;
      const float4 vv1 = *(const float4*)(Vb + (size_t)(t * 16 + r + 8) * Dh + c);
      VTs[(c + 0) * LV + r + 8] = vv1.x;
      VTs[(c + 1) * LV + r + 8] = vv1.y;
      VTs[(c + 2) * LV + r + 8] = vv1.z;
      VTs[(c + 3) * LV + r + 8] = vv1.w;
    }
    __syncthreads();

    // scores = Q(16x64) x K^T(64x16): B operand b[j] = K[n=lane%16][4j+koff+{0,1}]
    v8f sc = {};
#pragma unroll
    for (int j = 0; j < 16; ++j) {
      v2f b = *(const v2f*)&Ks[lr * LK + j * 4 + koff];
      sc = WMMA_F32(qa[j], b, sc);
    }

    // mask + online softmax
    const int kcol = t * 16 + lr;
    float sm[8], tmax[8];
#pragma unroll
    for (int i = 0; i < 8; ++i) {
      const int qpos = q0w + i + (hi << 3);
      bool ok;
      if (t == 0) {
        ok = (kcol <= qpos);                       // global prefix, causal
      } else {
        const int diff = qpos - kcol;
        ok = (diff >= 0) && (diff < WND);          // sliding window (k>=16 via t>=1)
      }
      const float s = sc[i] * 0.125f;              // 1/sqrt(64)
      sm[i]   = ok ? s : -1e30f;
      tmax[i] = sm[i];
    }
#pragma unroll
    for (int i = 0; i < 8; ++i) {  // row max across the 16 lanes holding this row
      float v = tmax[i];
      v = fmaxf(v, __shfl_xor(v, 1));
      v = fmaxf(v, __shfl_xor(v, 2));
      v = fmaxf(v, __shfl_xor(v, 4));
      v = fmaxf(v, __shfl_xor(v, 8));
      tmax[i] = v;
    }
#pragma unroll
    for (int i = 0; i < 8; ++i) {
      const float mnew  = fmaxf(mrow[i], tmax[i]);
      const float alpha = __expf(mrow[i] - mnew);
      mrow[i] = mnew;
      const float p = (sm[i] > -1e29f) ? __expf(sm[i] - mnew) : 0.0f;
      float rs = p;  // row sum
      rs += __shfl_xor(rs, 1);
      rs += __shfl_xor(rs, 2);
      rs += __shfl_xor(rs, 4);
      rs += __shfl_xor(rs, 8);
      lrow[i] = lrow[i] * alpha + rs;
      acc0[i] *= alpha; acc1[i] *= alpha; acc2[i] *= alpha; acc3[i] *= alpha;
      Ps[wave][(i + (hi << 3)) * LV + lr] = p;     // C-layout -> LDS for A-operand reuse
    }
    __syncthreads();

    // O += P(16x16) x V(16x64): A from Ps, B operand b = V[k][d] = VTs[d][k..k+1]
#pragma unroll
    for (int j = 0; j < 4; ++j) {
      v2f a  = *(const v2f*)&Ps[wave][lr * LV + j * 4 + koff];
      v2f b0 = *(const v2f*)&VTs[(0 * 16 + lr) * LV + j * 4 + koff];
      acc0 = WMMA_F32(a, b0, acc0);
      v2f b1 = *(const v2f*)&VTs[(1 * 16 + lr) * LV + j * 4 + koff];
      acc1 = WMMA_F32(a, b1, acc1);
      v2f b2 = *(const v2f*)&VTs[(2 * 16 + lr) * LV + j * 4 + koff];
      acc2 = WMMA_F32(a, b2, acc2);
      v2f b3 = *(const v2f*)&VTs[(3 * 16 + lr) * LV + j * 4 + koff];
      acc3 = WMMA_F32(a, b3, acc3);
    }
  }

  // finalize: O stored head-interleaved [B, S, H*Dh] for the output projection
  const int bq = bh >> 4;
  const int hq = bh & 15;
#pragma unroll
  for (int i = 0; i < 8; ++i) {
    const float inv = 1.0f / lrow[i];
    const int qpos  = q0w + i + (hi << 3);
    float* orow = Og + (size_t)(bq * S_ + qpos) * (H * Dh) + hq * Dh;
    orow[ 0 + lr] = acc0[i] * inv;
    orow[16 + lr] = acc1[i] * inv;
    orow[32 + lr] = acc2[i] * inv;
    orow[48 + lr] = acc3[i] * inv;
  }
}

// ---------------------------------------------------------------------------
extern "C" void kernel_launch(void* const* d_in, const int* in_sizes, int n_in,
                              void* d_out, int out_size, void* d_ws, size_t ws_size,
                              hipStream_t stream) {
  (void)in_sizes; (void)n_in; (void)out_size; (void)ws_size;

  const float* x     = (const float*)d_in[0];   // [2,4096,1024]
  const float* w_qkv = (const float*)d_in[1];   // [3072,1024]
  const float* w_out = (const float*)d_in[2];   // [1024,1024]
  float* out = (float*)d_out;                   // [2,4096,1024]
  float* ws  = (float*)d_ws;

  const size_t per = 2ull * 16 * 4096 * 64;     // 8388608 floats per Q/K/V tensor
  float* Qb = ws;
  float* Kb = ws + per;
  float* Vb = ws + 2 * per;
  float* Ob = ws + 3 * per;                     // [2,4096,1024] attention output

  // 1) QKV projection, scattered to [Q|K|V][B,H,S,64]
  gemm_nt_wmma<1><<<dim3(3072 / 64, 8192 / 128), 256, 0, stream>>>(
      x, w_qkv, Qb, 8192, 3072, 1024);

  // 2) windowed + global-prefix causal attention
  attn_kernel<<<dim3(2 * 16 * (4096 / 64)), 128, 0, stream>>>(Qb, Kb, Vb, Ob);

  // 3) output projection
  gemm_nt_wmma<0><<<dim3(1024 / 64, 8192 / 128), 256, 0, stream>>>(
      Ob, w_out, out, 8192, 1024, 1024);
}